// Inflate_40845138985508
// MI455X (gfx1250) — compile-verified
//
#include <hip/hip_runtime.h>
#include <stdint.h>

// Ragged unfold (W=5, pad=2): out[n, d*5 + w] = x[row(n,w), d] or 0.
// Pure data movement: ~400 MB HBM traffic -> ~17us floor at 23.3 TB/s.
// CDNA5 path: GLOBAL_LOAD_ASYNC_TO_LDS_B128 (ASYNCcnt) staging + coalesced b128 stores.

#define WIN   5
#define HALF  2
#define DIM   1024
#define NSEQ  8
#define TPB   256   // 8 waves (wave32); each thread owns 4 consecutive d

#if defined(__has_builtin)
#if __has_builtin(__builtin_amdgcn_global_load_async_to_lds_b128)
#define USE_ASYNC_LDS 1
#endif
#endif
#ifndef USE_ASYNC_LDS
#define USE_ASYNC_LDS 0
#endif

typedef float __attribute__((ext_vector_type(4))) f4;

// Exact pointee type from the builtin's signature (GCC-style vector, per diagnostic):
typedef int v4i_g __attribute__((vector_size(16)));
typedef v4i_g __attribute__((address_space(1)))* async_gp;  // global (AS1)
typedef v4i_g __attribute__((address_space(3)))* async_lp;  // LDS (AS3)

__global__ __launch_bounds__(TPB) void inflate_unfold_kernel(
    const float* __restrict__ x,
    const int* __restrict__ lengths,
    float* __restrict__ out) {
  const int n = blockIdx.x;     // token id
  const int t = threadIdx.x;    // d-block id: d = 4t..4t+3

  // --- uniform segment search over 8 lengths (lowers to SALU/SMEM) ---
  int start = 0, len = 0, acc = 0;
#pragma unroll
  for (int s = 0; s < NSEQ; ++s) {
    int L = lengths[s];
    if (n >= acc && n < acc + L) { start = acc; len = L; }
    acc += L;
  }
  const int local = n - start;

  f4 v[WIN];

#if USE_ASYNC_LDS
  __shared__ float lds[WIN * DIM];   // 20 KB of the 320 KB/WGP pool
#pragma unroll
  for (int w = 0; w < WIN; ++w) {
    const int sl = local + w - HALF;
    const bool valid = (sl >= 0) && (sl < len);   // uniform across block
    if (valid) {
      const float* g = x + (size_t)(start + sl) * DIM + 4 * t;
      __builtin_amdgcn_global_load_async_to_lds_b128(
          (async_gp)g, (async_lp)&lds[w * DIM + 4 * t], 0, 0);
    } else {
      *(f4*)&lds[w * DIM + 4 * t] = (f4)0.0f;     // ds_store_b128 zero row
    }
  }
#if __has_builtin(__builtin_amdgcn_s_wait_asynccnt)
  __builtin_amdgcn_s_wait_asynccnt(0);
#else
  asm volatile("s_wait_asynccnt 0" ::: "memory");
#endif
  __syncthreads();
#pragma unroll
  for (int w = 0; w < WIN; ++w)
    v[w] = *(const f4*)&lds[w * DIM + 4 * t];     // ds_load_b128, conflict-free
#else
  // Fallback: direct register gather (still coalesced b128 reads)
#pragma unroll
  for (int w = 0; w < WIN; ++w) {
    const int sl = local + w - HALF;
    const bool valid = (sl >= 0) && (sl < len);
    v[w] = valid ? *(const f4*)(x + (size_t)(start + sl) * DIM + 4 * t)
                 : (f4)0.0f;
  }
#endif

  // --- interleave 20 outputs: flat j in [0,20), value = v[j%5][j/5] ---
  float elem[WIN * 4];
#pragma unroll
  for (int j = 0; j < WIN * 4; ++j) elem[j] = v[j % WIN][j / WIN];

  float* op = out + (size_t)n * (DIM * WIN) + (size_t)t * 20;  // byte 80*t: 16B aligned
#pragma unroll
  for (int k = 0; k < 5; ++k) {
    f4 o;
    o.x = elem[4 * k + 0];
    o.y = elem[4 * k + 1];
    o.z = elem[4 * k + 2];
    o.w = elem[4 * k + 3];
    *(f4*)(op + 4 * k) = o;   // global_store_b128, contiguous across wave
  }
}

extern "C" void kernel_launch(void* const* d_in, const int* in_sizes, int n_in,
                              void* d_out, int out_size, void* d_ws, size_t ws_size,
                              hipStream_t stream) {
  const float* x = (const float*)d_in[0];       // [N, 1024] fp32
  const int* lengths = (const int*)d_in[1];     // [8] int32
  float* out = (float*)d_out;                   // [N, 5120] fp32
  const int N = in_sizes[0] / DIM;              // 16384
  inflate_unfold_kernel<<<N, TPB, 0, stream>>>(x, lengths, out);
}